// HierarchicalEncoder_86346022519259
// MI455X (gfx1250) — compile-verified
//
#include <hip/hip_runtime.h>

#define NPG 32
#define PMAX 8

typedef __attribute__((ext_vector_type(2))) float v2f;
typedef __attribute__((ext_vector_type(8))) float v8f;

__device__ __forceinline__ float reluf(float v) { return v > 0.f ? v : 0.f; }

// ---------------------------------------------------------------------------
// h0 = relu(x @ W_in + b_in), node_types = argmax(x)
// ---------------------------------------------------------------------------
__global__ __launch_bounds__(256) void init_kernel(
    const float* __restrict__ x, const float* __restrict__ Win,
    const float* __restrict__ bin, float* __restrict__ h0,
    int* __restrict__ ntype, int N)
{
    int i = blockIdx.x * blockDim.x + threadIdx.x;
    if (i >= N) return;
    float x0 = x[i * 4 + 0], x1 = x[i * 4 + 1], x2 = x[i * 4 + 2], x3 = x[i * 4 + 3];
    int t = 0; float m = x0;
    if (x1 > m) { m = x1; t = 1; }
    if (x2 > m) { m = x2; t = 2; }
    if (x3 > m) { m = x3; t = 3; }
    ntype[i] = t;
    float* ho = h0 + (size_t)i * 64;
#pragma unroll 4
    for (int c = 0; c < 64; ++c) {
        float v = bin[c] + x0 * Win[c] + x1 * Win[64 + c] + x2 * Win[128 + c] + x3 * Win[192 + c];
        ho[c] = reluf(v);
    }
}

// ---------------------------------------------------------------------------
// msg = relu((h[src] + edge_attr@We + be) @ Wm + bm); atomicAdd into agg[dst]
// 4 waves/block; wave w owns output cols [16w,16w+16); 128 edges per block.
// ---------------------------------------------------------------------------
__global__ __launch_bounds__(128) void gnn_msg_kernel(
    const float* __restrict__ h, const float* __restrict__ edge_attr,
    const int* __restrict__ src, const int* __restrict__ dst,
    const float* __restrict__ We, const float* __restrict__ be,
    const float* __restrict__ Wm, const float* __restrict__ bm,
    float* __restrict__ agg, int E)
{
    __shared__ float sWe[7 * 64];
    __shared__ float sbe[64];
    __shared__ float sA[16 * 66];

    const int tid = threadIdx.x;
    for (int i = tid; i < 7 * 64; i += 128) sWe[i] = We[i];
    if (tid < 64) sbe[tid] = be[tid];

    const int lane  = tid & 31;
    const int wv    = tid >> 5;
    const int nn    = wv * 16 + (lane & 15);
    const int khalf = (lane >> 4) << 1;   // 0 or 2

    // resident B fragments: 16 k-steps of W [64x64]
    float bw0[16], bw1[16];
#pragma unroll
    for (int kk = 0; kk < 16; ++kk) {
        bw0[kk] = Wm[(4 * kk + khalf + 0) * 64 + nn];
        bw1[kk] = Wm[(4 * kk + khalf + 1) * 64 + nn];
    }
    const float biasn = bm[nn];

    const int srow  = tid >> 3;
    const int scol  = (tid & 7) * 8;
    const int mbase = (lane >> 4) << 3;   // 0 or 8
    const int mrow  = lane & 15;

    for (int t = 0; t < 8; ++t) {
        const int ebase = blockIdx.x * 128 + t * 16;
        __syncthreads();
        {
            const int e = ebase + srow;
            if (e < E) {
                const int s = src[e];
                const float* hrow = h + (size_t)s * 64;
                const float* ea = edge_attr + (size_t)e * 7;
                float f0 = ea[0], f1 = ea[1], f2 = ea[2], f3 = ea[3],
                      f4 = ea[4], f5 = ea[5], f6 = ea[6];
#pragma unroll
                for (int j = 0; j < 8; ++j) {
                    const int c = scol + j;
                    float v = sbe[c] + hrow[c];
                    v += f0 * sWe[0 * 64 + c]; v += f1 * sWe[1 * 64 + c];
                    v += f2 * sWe[2 * 64 + c]; v += f3 * sWe[3 * 64 + c];
                    v += f4 * sWe[4 * 64 + c]; v += f5 * sWe[5 * 64 + c];
                    v += f6 * sWe[6 * 64 + c];
                    sA[srow * 66 + c] = v;
                }
            } else {
#pragma unroll
                for (int j = 0; j < 8; ++j) sA[srow * 66 + scol + j] = 0.f;
            }
        }
        __syncthreads();

        v8f acc = {};
#pragma unroll
        for (int kk = 0; kk < 16; ++kk) {
            const int kb = 4 * kk + khalf;
            v2f a = { sA[mrow * 66 + kb], sA[mrow * 66 + kb + 1] };
            v2f b = { bw0[kk], bw1[kk] };
            acc = __builtin_amdgcn_wmma_f32_16x16x4_f32(
                false, a, false, b, (short)0, acc, false, false);
        }

        if (ebase + 16 <= E) {
            // full tile fast path: no per-row guards; batch dst index loads
            const int4* dv = (const int4*)(dst + ebase + mbase);  // 32B aligned
            const int4 d0 = dv[0];
            const int4 d1 = dv[1];
            int di[8] = { d0.x, d0.y, d0.z, d0.w, d1.x, d1.y, d1.z, d1.w };
#pragma unroll
            for (int r = 0; r < 8; ++r) {
                float v = reluf(acc[r] + biasn);
                atomicAdd(&agg[(size_t)di[r] * 64 + nn], v);
            }
        } else {
#pragma unroll
            for (int r = 0; r < 8; ++r) {
                const int em = ebase + mbase + r;
                if (em < E) {
                    float v = reluf(acc[r] + biasn);
                    const int d = dst[em];
                    atomicAdd(&agg[(size_t)d * 64 + nn], v);
                }
            }
        }
    }
}

// ---------------------------------------------------------------------------
// h_out = relu(h @ Ws + agg + bs) ; 128 nodes per block
// ---------------------------------------------------------------------------
__global__ __launch_bounds__(128) void gnn_self_kernel(
    const float* __restrict__ h, const float* __restrict__ agg,
    const float* __restrict__ Ws, const float* __restrict__ bs,
    float* __restrict__ hout, int N)
{
    __shared__ float sA[16 * 66];

    const int tid   = threadIdx.x;
    const int lane  = tid & 31;
    const int wv    = tid >> 5;
    const int nn    = wv * 16 + (lane & 15);
    const int khalf = (lane >> 4) << 1;

    float bw0[16], bw1[16];
#pragma unroll
    for (int kk = 0; kk < 16; ++kk) {
        bw0[kk] = Ws[(4 * kk + khalf + 0) * 64 + nn];
        bw1[kk] = Ws[(4 * kk + khalf + 1) * 64 + nn];
    }
    const float biasn = bs[nn];

    const int srow  = tid >> 3;
    const int scol  = (tid & 7) * 8;
    const int mbase = (lane >> 4) << 3;
    const int mrow  = lane & 15;

    for (int t = 0; t < 8; ++t) {
        const int nbase = blockIdx.x * 128 + t * 16;
        __syncthreads();
        {
            const int row = nbase + srow;
            if (row < N) {
                const float* hrow = h + (size_t)row * 64;
#pragma unroll
                for (int j = 0; j < 8; ++j) sA[srow * 66 + scol + j] = hrow[scol + j];
            } else {
#pragma unroll
                for (int j = 0; j < 8; ++j) sA[srow * 66 + scol + j] = 0.f;
            }
        }
        __syncthreads();

        v8f acc = {};
#pragma unroll
        for (int kk = 0; kk < 16; ++kk) {
            const int kb = 4 * kk + khalf;
            v2f a = { sA[mrow * 66 + kb], sA[mrow * 66 + kb + 1] };
            v2f b = { bw0[kk], bw1[kk] };
            acc = __builtin_amdgcn_wmma_f32_16x16x4_f32(
                false, a, false, b, (short)0, acc, false, false);
        }

        if (nbase + 16 <= N) {
            // full tile fast path: strided unguarded loads/stores
            const float* arow = agg  + (size_t)(nbase + mbase) * 64 + nn;
            float*       orow = hout + (size_t)(nbase + mbase) * 64 + nn;
#pragma unroll
            for (int r = 0; r < 8; ++r) {
                float v = acc[r] + biasn + arow[(size_t)r * 64];
                orow[(size_t)r * 64] = reluf(v);
            }
        } else {
#pragma unroll
            for (int r = 0; r < 8; ++r) {
                const int nd = nbase + mbase + r;
                if (nd < N) {
                    float v = acc[r] + biasn + agg[(size_t)nd * 64 + nn];
                    hout[(size_t)nd * 64 + nn] = reluf(v);
                }
            }
        }
    }
}

// ---------------------------------------------------------------------------
// per-graph mean + max pooling (32 nodes, 64 dims) -> pooled[B,128]
// ---------------------------------------------------------------------------
__global__ __launch_bounds__(64) void pool_kernel(
    const float* __restrict__ h, float* __restrict__ pooled)
{
    const int g = blockIdx.x;
    const int d = threadIdx.x;
    const float* base = h + (size_t)g * NPG * 64;
    float s = 0.f, mx = -3.402823466e38f;
    for (int i = 0; i < NPG; ++i) {
        float v = base[i * 64 + d];
        s += v;
        mx = fmaxf(mx, v);
    }
    pooled[(size_t)g * 128 + d]      = s * (1.f / NPG);
    pooled[(size_t)g * 128 + 64 + d] = mx;
}

// ---------------------------------------------------------------------------
// topo head: [128]->relu64->relu32->{mu2,lv2}; writes out cols 0..1
// ---------------------------------------------------------------------------
__global__ __launch_bounds__(64) void topo_kernel(
    const float* __restrict__ pooled,
    const float* __restrict__ W1, const float* __restrict__ b1,
    const float* __restrict__ W2, const float* __restrict__ b2,
    const float* __restrict__ Wmu, const float* __restrict__ bmu,
    const float* __restrict__ Wlv, const float* __restrict__ blv,
    float* __restrict__ out, int B)
{
    __shared__ float sin[128];
    __shared__ float st1[64];
    __shared__ float st2[32];
    const int g = blockIdx.x;
    const int t = threadIdx.x;
    sin[t]      = pooled[(size_t)g * 128 + t];
    sin[t + 64] = pooled[(size_t)g * 128 + 64 + t];
    __syncthreads();
    float a = b1[t];
    for (int k = 0; k < 128; ++k) a += sin[k] * W1[k * 64 + t];
    st1[t] = reluf(a);
    __syncthreads();
    if (t < 32) {
        float a2 = b2[t];
        for (int k = 0; k < 64; ++k) a2 += st1[k] * W2[k * 32 + t];
        st2[t] = reluf(a2);
    }
    __syncthreads();
    if (t < 4) {
        const int col = t & 1;
        const float* W = (t < 2) ? Wmu : Wlv;
        const float* bb = (t < 2) ? bmu : blv;
        float a3 = bb[col];
        for (int k = 0; k < 32; ++k) a3 += st2[k] * W[k * 2 + col];
        const size_t B8 = (size_t)B * 8;
        if (t < 2) {
            out[(size_t)g * 8 + col] = a3;            // z
            out[B8 + (size_t)g * 8 + col] = a3;       // mu
        } else {
            out[2 * B8 + (size_t)g * 8 + col] = a3;   // lv
        }
    }
}

// ---------------------------------------------------------------------------
// edge routing: last-edge indices for gv/vv (atomicMax), ot-encoder sum
// ---------------------------------------------------------------------------
__global__ __launch_bounds__(256) void route_kernel(
    const float* __restrict__ edge_attr,
    const int* __restrict__ src, const int* __restrict__ dst,
    const int* __restrict__ ntype, const int* __restrict__ batch,
    const float* __restrict__ Wot1, const float* __restrict__ bot1,
    const float* __restrict__ Wot2, const float* __restrict__ bot2,
    int* __restrict__ idx_gv, int* __restrict__ idx_vv,
    float* __restrict__ h_ot, int E)
{
    int e = blockIdx.x * blockDim.x + threadIdx.x;
    if (e >= E) return;
    const int s = src[e], d = dst[e];
    const int st = ntype[s], dt = ntype[d];
    const int lo = min(st, dt), hi = max(st, dt);
    const int g = batch[s];
    const bool gv = (lo == 0) && (hi == 2);
    const bool vv = (lo == 1) && (hi == 2);
    if (gv) {
        atomicMax(&idx_gv[g], e);
    } else if (vv) {
        atomicMax(&idx_vv[g], e);
    } else {
        const float* ea = edge_attr + (size_t)e * 7;
        float f[7];
#pragma unroll
        for (int q = 0; q < 7; ++q) f[q] = ea[q];
        float hid[32];
#pragma unroll 4
        for (int c = 0; c < 32; ++c) {
            float v = bot1[c];
#pragma unroll
            for (int q = 0; q < 7; ++q) v += f[q] * Wot1[q * 32 + c];
            hid[c] = reluf(v);
        }
#pragma unroll 2
        for (int j = 0; j < 16; ++j) {
            float o = bot2[j];
#pragma unroll 8
            for (int c = 0; c < 32; ++c) o += hid[c] * Wot2[c * 16 + j];
            atomicAdd(&h_ot[(size_t)g * 16 + j], o);
        }
    }
}

// ---------------------------------------------------------------------------
// gv/vv last-edge encoders: threads 0-15 -> gv, 16-31 -> vv
// ---------------------------------------------------------------------------
__global__ __launch_bounds__(32) void lastenc_kernel(
    const float* __restrict__ edge_attr,
    const int* __restrict__ idx_gv, const int* __restrict__ idx_vv,
    const float* __restrict__ Wg1, const float* __restrict__ bg1,
    const float* __restrict__ Wg2, const float* __restrict__ bg2,
    const float* __restrict__ Wv1, const float* __restrict__ bv1,
    const float* __restrict__ Wv2, const float* __restrict__ bv2,
    float* __restrict__ h_gv, float* __restrict__ h_vv)
{
    const int g = blockIdx.x;
    const int t = threadIdx.x;
    const bool isGv = (t < 16);
    const int j = t & 15;
    const int idx = isGv ? idx_gv[g] : idx_vv[g];
    const float* W1 = isGv ? Wg1 : Wv1;
    const float* b1 = isGv ? bg1 : bv1;
    const float* W2 = isGv ? Wg2 : Wv2;
    const float* b2 = isGv ? bg2 : bv2;
    const int ci = idx < 0 ? 0 : idx;
    const float* ea = edge_attr + (size_t)ci * 7;
    float f[7];
#pragma unroll
    for (int q = 0; q < 7; ++q) f[q] = ea[q];
    float o = b2[j];
    for (int c = 0; c < 32; ++c) {
        float hv = b1[c];
#pragma unroll
        for (int q = 0; q < 7; ++q) hv += f[q] * W1[q * 32 + c];
        hv = reluf(hv);
        o += hv * W2[c * 16 + j];
    }
    o = (idx >= 0) ? o : 0.f;
    if (isGv) h_gv[(size_t)g * 16 + j] = o;
    else      h_vv[(size_t)g * 16 + j] = o;
}

// ---------------------------------------------------------------------------
// value head: concat(h_gv,h_vv,h_ot)[48] -> relu32 -> {mu2,lv2}; cols 2..3
// ---------------------------------------------------------------------------
__global__ __launch_bounds__(256) void value_kernel(
    const float* __restrict__ h_gv, const float* __restrict__ h_vv,
    const float* __restrict__ h_ot,
    const float* __restrict__ Wvc, const float* __restrict__ bvc,
    const float* __restrict__ Wmu, const float* __restrict__ bmu,
    const float* __restrict__ Wlv, const float* __restrict__ blv,
    float* __restrict__ out, int B)
{
    int g = blockIdx.x * blockDim.x + threadIdx.x;
    if (g >= B) return;
    float c[48];
#pragma unroll
    for (int k = 0; k < 16; ++k) {
        c[k]      = h_gv[(size_t)g * 16 + k];
        c[16 + k] = h_vv[(size_t)g * 16 + k];
        c[32 + k] = h_ot[(size_t)g * 16 + k];
    }
    float v[32];
#pragma unroll 4
    for (int j = 0; j < 32; ++j) {
        float a = bvc[j];
#pragma unroll 8
        for (int k = 0; k < 48; ++k) a += c[k] * Wvc[k * 32 + j];
        v[j] = reluf(a);
    }
    const size_t B8 = (size_t)B * 8;
#pragma unroll
    for (int col = 0; col < 2; ++col) {
        float mu = bmu[col], lv = blv[col];
#pragma unroll 8
        for (int k = 0; k < 32; ++k) {
            mu += v[k] * Wmu[k * 2 + col];
            lv += v[k] * Wlv[k * 2 + col];
        }
        out[(size_t)g * 8 + 2 + col] = mu;           // z
        out[B8 + (size_t)g * 8 + 2 + col] = mu;      // mu
        out[2 * B8 + (size_t)g * 8 + 2 + col] = lv;  // lv
    }
}

// ---------------------------------------------------------------------------
// poles/zeros DeepSets + pz head; cols 4..7
// ---------------------------------------------------------------------------
__device__ __forceinline__ void dsets_branch(
    const float* __restrict__ pts,   // [PMAX,2] for this graph
    const float* __restrict__ Wp1, const float* __restrict__ bp1,
    const float* __restrict__ Wp2, const float* __restrict__ bp2,
    const float* __restrict__ Wr,  const float* __restrict__ br,
    float* __restrict__ outv)        // [16]
{
    float sum[32];
#pragma unroll
    for (int k = 0; k < 32; ++k) sum[k] = 0.f;
    for (int i = 0; i < PMAX; ++i) {
        const float pa = pts[i * 2 + 0];
        const float pb = pts[i * 2 + 1];
        float hid[32];
#pragma unroll 8
        for (int cidx = 0; cidx < 32; ++cidx)
            hid[cidx] = reluf(bp1[cidx] + pa * Wp1[cidx] + pb * Wp1[32 + cidx]);
#pragma unroll 2
        for (int c2 = 0; c2 < 32; ++c2) {
            float o = bp2[c2];
#pragma unroll 8
            for (int k = 0; k < 32; ++k) o += hid[k] * Wp2[k * 32 + c2];
            sum[c2] += o;
        }
    }
#pragma unroll 2
    for (int j = 0; j < 16; ++j) {
        float o = br[j];
#pragma unroll 8
        for (int k = 0; k < 32; ++k) o += sum[k] * Wr[k * 16 + j];
        outv[j] = o;
    }
}

__global__ __launch_bounds__(256) void pz_kernel(
    const float* __restrict__ poles, const float* __restrict__ zerosArr,
    const float* __restrict__ Wp1, const float* __restrict__ bp1,
    const float* __restrict__ Wp2, const float* __restrict__ bp2,
    const float* __restrict__ Wpr, const float* __restrict__ bpr,
    const float* __restrict__ Wz1, const float* __restrict__ bz1,
    const float* __restrict__ Wz2, const float* __restrict__ bz2,
    const float* __restrict__ Wzr, const float* __restrict__ bzr,
    const float* __restrict__ Wpzc, const float* __restrict__ bpzc,
    const float* __restrict__ Wmu, const float* __restrict__ bmu,
    const float* __restrict__ Wlv, const float* __restrict__ blv,
    float* __restrict__ out, int B)
{
    int g = blockIdx.x * blockDim.x + threadIdx.x;
    if (g >= B) return;
    float ds[32];
    dsets_branch(poles    + (size_t)g * PMAX * 2, Wp1, bp1, Wp2, bp2, Wpr, bpr, ds);
    dsets_branch(zerosArr + (size_t)g * PMAX * 2, Wz1, bz1, Wz2, bz2, Wzr, bzr, ds + 16);
    float pz[32];
#pragma unroll 2
    for (int j = 0; j < 32; ++j) {
        float a = bpzc[j];
#pragma unroll 8
        for (int k = 0; k < 32; ++k) a += ds[k] * Wpzc[k * 32 + j];
        pz[j] = reluf(a);
    }
    const size_t B8 = (size_t)B * 8;
#pragma unroll
    for (int col = 0; col < 4; ++col) {
        float mu = bmu[col], lv = blv[col];
#pragma unroll 8
        for (int k = 0; k < 32; ++k) {
            mu += pz[k] * Wmu[k * 4 + col];
            lv += pz[k] * Wlv[k * 4 + col];
        }
        out[(size_t)g * 8 + 4 + col] = mu;           // z
        out[B8 + (size_t)g * 8 + 4 + col] = mu;      // mu
        out[2 * B8 + (size_t)g * 8 + 4 + col] = lv;  // lv
    }
}

// ---------------------------------------------------------------------------
extern "C" void kernel_launch(void* const* d_in, const int* in_sizes, int n_in,
                              void* d_out, int out_size, void* d_ws, size_t ws_size,
                              hipStream_t stream)
{
    const float* x         = (const float*)d_in[0];
    const int*   eidx      = (const int*)d_in[1];
    const float* edge_attr = (const float*)d_in[2];
    const int*   batch     = (const int*)d_in[3];
    const float* poles     = (const float*)d_in[4];
    const float* zerosArr  = (const float*)d_in[5];

    const int N = in_sizes[0] / 4;
    const int E = in_sizes[2] / 7;
    const int B = in_sizes[4] / (PMAX * 2);

    const int* src = eidx;       // edge_index[0]
    const int* dst = eidx + E;   // edge_index[1]

    // params: dict insertion order from _make_params
    const float* P[60];
    for (int i = 0; i < 60; ++i) P[i] = (const float*)d_in[6 + i];
    const float *Win = P[0], *bin = P[1], *We = P[2], *be = P[3];
    const float *Wt1 = P[16], *bt1 = P[17], *Wt2 = P[18], *bt2 = P[19];
    const float *Wtmu = P[20], *btmu = P[21], *Wtlv = P[22], *btlv = P[23];
    const float *Wg1 = P[24], *bg1 = P[25], *Wg2 = P[26], *bg2 = P[27];
    const float *Wv1 = P[28], *bv1 = P[29], *Wv2 = P[30], *bv2 = P[31];
    const float *Wo1 = P[32], *bo1 = P[33], *Wo2 = P[34], *bo2 = P[35];
    const float *Wvc = P[36], *bvc = P[37], *Wvmu = P[38], *bvmu = P[39];
    const float *Wvlv = P[40], *bvlv = P[41];
    const float *Wp1 = P[42], *bp1 = P[43], *Wp2 = P[44], *bp2 = P[45];
    const float *Wpr = P[46], *bpr = P[47];
    const float *Wz1 = P[48], *bz1 = P[49], *Wz2 = P[50], *bz2 = P[51];
    const float *Wzr = P[52], *bzr = P[53];
    const float *Wpzc = P[54], *bpzc = P[55];
    const float *Wpmu = P[56], *bpmu = P[57], *Wplv = P[58], *bplv = P[59];

    // workspace layout
    float* hA     = (float*)d_ws;
    float* hB     = hA + (size_t)N * 64;
    float* agg    = hB + (size_t)N * 64;
    float* pooled = agg + (size_t)N * 64;
    int*   ntype  = (int*)(pooled + (size_t)B * 128);
    int*   idx_gv = ntype + N;
    int*   idx_vv = idx_gv + B;
    float* h_gv   = (float*)(idx_vv + B);
    float* h_vv   = h_gv + (size_t)B * 16;
    float* h_ot   = h_vv + (size_t)B * 16;
    float* out    = (float*)d_out;

    // Stage 0: init
    init_kernel<<<(N + 255) / 256, 256, 0, stream>>>(x, Win, bin, hA, ntype, N);

    // Stage 1: 3-layer GNN (ping-pong hA/hB)
    float* hcur = hA;
    float* hnext = hB;
    const int msgGrid  = (E + 127) / 128;
    const int selfGrid = (N + 127) / 128;
    for (int l = 0; l < 3; ++l) {
        const float* Wm = P[4 + 4 * l];
        const float* bm = P[5 + 4 * l];
        const float* Ws = P[6 + 4 * l];
        const float* bs = P[7 + 4 * l];
        hipMemsetAsync(agg, 0, (size_t)N * 64 * sizeof(float), stream);
        gnn_msg_kernel<<<msgGrid, 128, 0, stream>>>(
            hcur, edge_attr, src, dst, We, be, Wm, bm, agg, E);
        gnn_self_kernel<<<selfGrid, 128, 0, stream>>>(
            hcur, agg, Ws, bs, hnext, N);
        float* tmp = hcur; hcur = hnext; hnext = tmp;
    }

    // Branch 1: topology
    pool_kernel<<<B, 64, 0, stream>>>(hcur, pooled);
    topo_kernel<<<B, 64, 0, stream>>>(pooled, Wt1, bt1, Wt2, bt2,
                                      Wtmu, btmu, Wtlv, btlv, out, B);

    // Branch 2: values
    hipMemsetAsync(idx_gv, 0xFF, (size_t)B * sizeof(int), stream);
    hipMemsetAsync(idx_vv, 0xFF, (size_t)B * sizeof(int), stream);
    hipMemsetAsync(h_ot, 0, (size_t)B * 16 * sizeof(float), stream);
    route_kernel<<<(E + 255) / 256, 256, 0, stream>>>(
        edge_attr, src, dst, ntype, batch, Wo1, bo1, Wo2, bo2,
        idx_gv, idx_vv, h_ot, E);
    lastenc_kernel<<<B, 32, 0, stream>>>(
        edge_attr, idx_gv, idx_vv, Wg1, bg1, Wg2, bg2,
        Wv1, bv1, Wv2, bv2, h_gv, h_vv);
    value_kernel<<<(B + 255) / 256, 256, 0, stream>>>(
        h_gv, h_vv, h_ot, Wvc, bvc, Wvmu, bvmu, Wvlv, bvlv, out, B);

    // Branch 3: poles/zeros DeepSets
    pz_kernel<<<(B + 255) / 256, 256, 0, stream>>>(
        poles, zerosArr, Wp1, bp1, Wp2, bp2, Wpr, bpr,
        Wz1, bz1, Wz2, bz2, Wzr, bzr,
        Wpzc, bpzc, Wpmu, bpmu, Wplv, bplv, out, B);
}